// CosineLoss_90374701842716
// MI455X (gfx1250) — compile-verified
//
#include <hip/hip_runtime.h>
#include <hip/hip_bf16.h>

// CosineLoss gather-reduction for MI455X (gfx1250, wave32).
//
// loss = mean_i (1 - output[i, targets[i]])
//   output : (8192, 32000) fp32, targets : (8192,) int
//
// Latency-bound gather (~1 MB of distinct cachelines out of a 1 GB matrix).
// Wave-level reduction is done with V_WMMA_F32_16X16X4_F32 (B = ones), which
// collapses 32 fp32 lane partials in a single matrix op, fp32-exact.

typedef __attribute__((ext_vector_type(2))) float v2f;
typedef __attribute__((ext_vector_type(8))) float v8f;

#define NUM_CLASSES 32000
#define N_ROWS      8192
#define BLOCK       256
#define NBLOCKS     (N_ROWS / BLOCK)   // 32

// Full-wave (32 lane) fp32 sum; result valid in every lane.
__device__ __forceinline__ float wave_sum_f32(float p) {
#if defined(__gfx1250__) && __has_builtin(__builtin_amdgcn_wmma_f32_16x16x4_f32)
    // A is 16x4 fp32: lane L holds A[L%16][ L<16 ? {0,1} : {2,3} ] in 2 VGPRs.
    // Put the lane partial in a[0], zero in a[1]  ->  A[m][0]=p[m], A[m][2]=p[m+16].
    v2f a; a[0] = p;    a[1] = 0.0f;
    // B is 4x16 all-ones -> D[m][n] = sum_k A[m][k] = p[m] + p[m+16], all n equal.
    v2f b; b[0] = 1.0f; b[1] = 1.0f;
    v8f c = {};
    v8f d = __builtin_amdgcn_wmma_f32_16x16x4_f32(
        /*neg_a=*/false, a, /*neg_b=*/false, b,
        /*c_mod=*/(short)0, c, /*reuse_a=*/false, /*reuse_b=*/false);
    // D layout: VGPR j, lanes 0-15 -> M=j ; lanes 16-31 -> M=8+j.
    // Sum of this lane's 8 D registers = half-matrix sum (rows 0-7 or 8-15).
    float t = d[0] + d[1] + d[2] + d[3] + d[4] + d[5] + d[6] + d[7];
    // Combine the two halves (lanes 0-15 hold M=0..7 total, 16-31 hold M=8..15).
    t += __shfl_xor(t, 16, 32);
    return t;
#else
    // Fallback: shuffle ladder (kept only so the file always compiles).
    #pragma unroll
    for (int off = 16; off > 0; off >>= 1) p += __shfl_xor(p, off, 32);
    return p;
#endif
}

__global__ void __launch_bounds__(BLOCK)
cosine_gather_partials(const float* __restrict__ output,
                       const int*   __restrict__ targets,
                       float*       __restrict__ partials) {
    const int i = blockIdx.x * BLOCK + threadIdx.x;   // one row per thread; N % BLOCK == 0
    // Gather: each element is on its own 128B line; NT hint -> don't pollute L2.
    const int    t   = targets[i];
    const size_t idx = (size_t)i * (size_t)NUM_CLASSES + (size_t)t;
    const float  g   = __builtin_nontemporal_load(output + idx);
    const float  p   = 1.0f - g;

    // Wave32 reduction via fp32 WMMA (EXEC is all-ones here; no divergence).
    const float wsum = wave_sum_f32(p);

    __shared__ float lds[BLOCK / 32];
    const int lane = threadIdx.x & 31;
    const int wid  = threadIdx.x >> 5;
    if (lane == 0) lds[wid] = wsum;
    __syncthreads();

    if (threadIdx.x == 0) {
        float s = 0.0f;
        #pragma unroll
        for (int k = 0; k < BLOCK / 32; ++k) s += lds[k];
        partials[blockIdx.x] = s;   // plain store: deterministic across graph replays
    }
}

__global__ void __launch_bounds__(32)
cosine_finalize(const float* __restrict__ partials,
                float*       __restrict__ out) {
    // One full wave: gather the 32 block partials (NBLOCKS == 32) and reduce.
    float p = 0.0f;
    for (int k = threadIdx.x; k < NBLOCKS; k += 32) p += partials[k];
    const float s = wave_sum_f32(p);
    if (threadIdx.x == 0) out[0] = s * (1.0f / (float)N_ROWS);
}

extern "C" void kernel_launch(void* const* d_in, const int* in_sizes, int n_in,
                              void* d_out, int out_size, void* d_ws, size_t ws_size,
                              hipStream_t stream) {
    const float* output  = (const float*)d_in[0];   // (8192, 32000) fp32
    const int*   targets = (const int*)d_in[1];     // (8192,) int
    float*       result  = (float*)d_out;           // scalar fp32
    float*       partials = (float*)d_ws;           // NBLOCKS floats of scratch

    cosine_gather_partials<<<NBLOCKS, BLOCK, 0, stream>>>(output, targets, partials);
    cosine_finalize<<<1, 32, 0, stream>>>(partials, result);
}